// QAEModel_31834297598676
// MI455X (gfx1250) — compile-verified
//
#include <hip/hip_runtime.h>
#include <hip/hip_bf16.h>

// ---------------------------------------------------------------------------
// 20-qubit QAE circuit simulator for MI455X (gfx1250, wave32).
//
// State: 8 x 2^20 f32 (32 MB) -> L2-resident (192 MB L2).  Per layer:
//   RY layer  = Kron of 20 RY(2x2)  -> five 16x16 fp32 group-gates,
//               applied with chained V_WMMA_F32_16X16X4_F32 (full precision).
//   ring CNOT = GF(2)-linear basis permutation; its inverse in flat-index
//               bits is a Gray-code map  a = (c ^ (c>>1)) ^ ((c&1)*0xC0000),
//               fused as a gather into the next layer's LDS tile load.
// Bulk LDS<->global tile traffic uses the Tensor Data Mover (TDM):
// tensor_load_to_lds / tensor_store_from_lds, one descriptor per wave
// (wave-uniform issue; EXEC is ignored by TDM so no divergent-issue hazard),
// synchronized with s_wait_tensorcnt + workgroup barrier.
// ---------------------------------------------------------------------------

typedef __attribute__((ext_vector_type(2))) float v2f;
typedef __attribute__((ext_vector_type(8))) float v8f;
typedef unsigned int u32;
typedef u32 u32x4 __attribute__((ext_vector_type(4)));
typedef int i32x4 __attribute__((ext_vector_type(4)));
typedef int i32x8 __attribute__((ext_vector_type(8)));

#define NV        20        // variational qubits
#define NLAYERS   12        // entangling layers (13th RY layer has no CNOT)
#define BATCH     8
#define TILE_ELEMS 65536    // 2^16 floats = 256 KB LDS tile (WGP has 320 KB)

// ---------------- TDM descriptor builder (ISA cdna5 ch.8 D# layout) ---------
// 2D tile: td1 rows of td0 4-byte elements; global row stride = stride0 elems;
// LDS rows packed contiguously starting at lds_byte.
__device__ __forceinline__ void tdm_desc(u32x4& g0, i32x8& g1, const void* ga_,
                                         u32 lds_byte, u32 td0, u32 td1,
                                         u32 stride0) {
  const unsigned long long ga = (unsigned long long)ga_;
  g0[0] = 1u;                                   // count=1, user descriptor
  g0[1] = lds_byte;                             // lds_addr (bytes)
  g0[2] = (u32)ga;                              // global_addr[31:0]
  g0[3] = ((u32)(ga >> 32) & 0x1FFFFFFu) | (2u << 30);  // ga[56:32] | type=2
  g1[0] = (int)(2u << 16);                      // wg_mask=0, data_size=2 (4B)
  g1[1] = (int)((td0 & 0xFFFFu) << 16);         // abar=0 | tensor_dim0.lo
  g1[2] = (int)((td0 >> 16) | ((td1 & 0xFFFFu) << 16)); // td0.hi | tensor_dim1.lo
  g1[3] = (int)((td1 >> 16) | ((td0 & 0xFFFFu) << 16)); // td1.hi | tile_dim0
  g1[4] = (int)(td1 & 0xFFFFu);                 // tile_dim1 | tile_dim2=0
  g1[5] = (int)stride0;                         // tensor_dim0_stride.lo32
  g1[6] = 0;                                    // stride0.hi16 | dim1_stride.lo
  g1[7] = 0;
}

__device__ __forceinline__ void tdm_load(const void* ga, u32 lds_byte, u32 td0,
                                         u32 td1, u32 stride0) {
  u32x4 g0; i32x8 g1;
  tdm_desc(g0, g1, ga, lds_byte, td0, td1, stride0);
  const i32x4 z4 = {0, 0, 0, 0};
  const i32x8 z8 = {0, 0, 0, 0, 0, 0, 0, 0};
  __builtin_amdgcn_tensor_load_to_lds(g0, g1, z4, z4, z8, 0);
}

__device__ __forceinline__ void tdm_store(const void* ga, u32 lds_byte, u32 td0,
                                          u32 td1, u32 stride0) {
  u32x4 g0; i32x8 g1;
  tdm_desc(g0, g1, ga, lds_byte, td0, td1, stride0);
  const i32x4 z4 = {0, 0, 0, 0};
  const i32x8 z8 = {0, 0, 0, 0, 0, 0, 0, 0};
  __builtin_amdgcn_tensor_store_from_lds(g0, g1, z4, z4, z8, 0);
}

// ---------------- 16x16x16 fp32 matmul: D = A(16x16) * X(16x16) -------------
// Chained as 4x V_WMMA_F32_16X16X4_F32 (K=4 per op).
__device__ __forceinline__ v8f wmma_16x16x16_f32(const v2f a[4], const v2f b[4]) {
  v8f acc = {0.f, 0.f, 0.f, 0.f, 0.f, 0.f, 0.f, 0.f};
#pragma unroll
  for (int kk = 0; kk < 4; ++kk) {
    acc = __builtin_amdgcn_wmma_f32_16x16x4_f32(
        /*neg_a=*/false, a[kk], /*neg_b=*/false, b[kk],
        /*c_mod=*/(short)0, acc, /*reuse_a=*/false, /*reuse_b=*/false);
  }
  return acc;
}

// Apply one 16x16 group-gate along the qubit axis of stride S inside a
// 65536-float LDS tile.  Tile index:  u = hi*(16*S) + k*S + lo.
// 256 disjoint jobs of (16 rows x 16 cols) -> in-place is safe.
// Wave-uniform job loop keeps EXEC all-ones around the WMMAs.
template <int S>
__device__ __forceinline__ void apply_group(float* __restrict__ tile,
                                            const float* __restrict__ AG,
                                            int lane, int wave) {
  // A-operand (16x4 f32 per chunk, 2 VGPRs): lanes 0-15 hold K={4kk,4kk+1},
  // lanes 16-31 hold K={4kk+2,4kk+3}; M = lane&15.  AG[m*16+k] = Gt[m][k].
  const int m    = lane & 15;
  const int koff = (lane >> 4) << 1;  // 0 or 2
  v2f a[4];
#pragma unroll
  for (int kk = 0; kk < 4; ++kk) {
    a[kk].x = AG[m * 16 + kk * 4 + koff];
    a[kk].y = AG[m * 16 + kk * 4 + koff + 1];
  }
  constexpr int rowstride = (S == 1) ? 1 : S;   // stride of the 16-axis (k)
  constexpr int colstride = (S == 1) ? 16 : 1;  // stride between columns (n)
  const int n = lane & 15;

  for (int j = wave; j < 256; j += 8) {         // 8 waves x 32 jobs
    int jobbase;
    if constexpr (S == 1) {
      jobbase = j << 8;                          // cols = 16 consecutive hi's
    } else {
      constexpr int colJobsPerHi = S >> 4;       // S/16 column-jobs per hi
      const int hi  = j / colJobsPerHi;
      const int lob = j - hi * colJobsPerHi;
      jobbase = hi * (S << 4) + (lob << 4);
    }
    const int cbase = jobbase + n * colstride;

    // B-operand (4x16 f32 per chunk): lanes 0-15 rows {4kk,4kk+1},
    // lanes 16-31 rows {4kk+2,4kk+3}; N = lane&15.
    v2f b[4];
#pragma unroll
    for (int kk = 0; kk < 4; ++kk) {
      const int k0 = kk * 4 + koff;
      b[kk].x = tile[cbase + k0 * rowstride];
      b[kk].y = tile[cbase + (k0 + 1) * rowstride];
    }

    const v8f d = wmma_16x16x16_f32(a, b);

    // C/D layout: VGPR r -> M=r (lanes 0-15) / M=r+8 (lanes 16-31), N=lane&15
    const int mbase = (lane >> 4) << 3;
#pragma unroll
    for (int r = 0; r < 8; ++r)
      tile[cbase + (mbase + r) * rowstride] = d[r];
  }
}

// ------------- gate-matrix prep: 13 layers x 5 groups of 16x16 --------------
// Per qubit (row-vector convention): new[j] = sum_i old[i]*G2[i][j],
// G2 = [[c,-s],[s,c]].  Stored transposed for the WMMA A-operand:
// AG[m*16+k] = prod_t G2[k_t][m_t]  (bit 3-t of k/m <-> qubit 4g+t).
extern "C" __global__ void qae_prep(const float* __restrict__ theta,
                                    float* __restrict__ AG) {
  const int blk   = blockIdx.x;      // layer*5 + g
  const int layer = blk / 5;
  const int g     = blk - layer * 5;
  const int tid   = threadIdx.x;     // m*16 + k
  const int m = tid >> 4, k = tid & 15;
  float prod = 1.f;
#pragma unroll
  for (int t = 0; t < 4; ++t) {
    const float th = 0.5f * theta[layer * NV + 4 * g + t];
    const float c = cosf(th), s = sinf(th);
    const int kb = (k >> (3 - t)) & 1;
    const int mb = (m >> (3 - t)) & 1;
    prod *= kb ? (mb ? c : s) : (mb ? -s : c);
  }
  AG[blk * 256 + tid] = prod;
}

// --- Pass A: contiguous 2^16 tile, optional fused CNOT-inverse gather, then
// --- qubit groups 1..4 (axis strides 4096 / 256 / 16 / 1) entirely in LDS.
extern "C" __global__ __launch_bounds__(256)
void qae_low(const float* __restrict__ src, float* __restrict__ dst,
             const float* __restrict__ AG5, int do_perm) {
  extern __shared__ float tile[];
  const int tid  = threadIdx.x;
  const int lane = tid & 31, wave = tid >> 5;
  const int b        = blockIdx.x >> 4;
  const unsigned base = (unsigned)(blockIdx.x & 15) << 16;   // top-4-bit slab
  const float* __restrict__ sb = src + ((size_t)b << 20);
  float* __restrict__       db = dst + ((size_t)b << 20);

  if (do_perm) {
    // out[c] = in[Pinv(c)];  Pinv(c) = (c ^ (c>>1)) ^ ((c&1) * 0xC0000).
    // Gray map preserves high bits -> each thread's gathers stay in 2 blocks.
#pragma unroll 4
    for (int i = 0; i < 256; ++i) {
      const unsigned u = ((unsigned)i << 8) + (unsigned)tid;
      const unsigned c = base | u;
      const unsigned a = (c ^ (c >> 1)) ^ ((c & 1u) ? 0xC0000u : 0u);
      tile[u] = sb[a];
    }
  } else {
    // Contiguous 256 KB stage-in: one 32 KB TDM descriptor per wave.
    tdm_load(sb + base + (unsigned)(wave << 13), (u32)(wave << 15),
             8192, 1, 8192);
    __builtin_amdgcn_s_wait_tensorcnt(0);
  }
  __syncthreads();
  apply_group<4096>(tile, AG5 + 1 * 256, lane, wave);  // qubits 4..7
  __syncthreads();
  apply_group<256>(tile, AG5 + 2 * 256, lane, wave);   // qubits 8..11
  __syncthreads();
  apply_group<16>(tile, AG5 + 3 * 256, lane, wave);    // qubits 12..15
  __syncthreads();
  apply_group<1>(tile, AG5 + 4 * 256, lane, wave);     // qubits 16..19
  __syncthreads();
  // Contiguous 256 KB stage-out via TDM (32 KB per wave).
  tdm_store(db + base + (unsigned)(wave << 13), (u32)(wave << 15),
            8192, 1, 8192);
  __builtin_amdgcn_s_wait_tensorcnt(0);
}

// --- Pass B: comb tile (16 rows of 4096 floats, row stride 2^16) handles
// --- qubit group 0 (the top 4 index bits); LDS row stride 4096.
// --- Row pairs are moved by per-wave TDM 2D descriptors.
extern "C" __global__ __launch_bounds__(256)
void qae_hi(const float* __restrict__ src, float* __restrict__ dst,
            const float* __restrict__ AG0) {
  extern __shared__ float tile[];
  const int tid  = threadIdx.x;
  const int lane = tid & 31, wave = tid >> 5;
  const int b       = blockIdx.x >> 4;
  const unsigned cb = blockIdx.x & 15;               // column block (4096 floats)
  const float* __restrict__ sb = src + ((size_t)b << 20);
  float* __restrict__       db = dst + ((size_t)b << 20);

  // Wave w stages rows {2w, 2w+1}: 2 x 4096 floats, global stride 2^16 elems.
  const unsigned row0 = (unsigned)(wave << 1);
  tdm_load(sb + (row0 << 16) + (cb << 12), (u32)(row0 << 14),
           4096, 2, 65536);
  __builtin_amdgcn_s_wait_tensorcnt(0);
  __syncthreads();
  apply_group<4096>(tile, AG0, lane, wave);          // qubits 0..3
  __syncthreads();
  tdm_store(db + (row0 << 16) + (cb << 12), (u32)(row0 << 14),
            4096, 2, 65536);
  __builtin_amdgcn_s_wait_tensorcnt(0);
}

// ---- probabilities over the low-3 qubits: out[b,j] = sum |amp|^2, j=idx&7 --
// Fixed-order LDS tree reduction: deterministic (no float atomics).
extern "C" __global__ __launch_bounds__(256)
void qae_probs(const float* __restrict__ s, float* __restrict__ out) {
  __shared__ float red[8 * 256];
  const int b = blockIdx.x, tid = threadIdx.x;
  const float* __restrict__ sb = s + ((size_t)b << 20);
  float acc[8];
#pragma unroll
  for (int j = 0; j < 8; ++j) acc[j] = 0.f;
  for (int g = tid; g < (1 << 17); g += 256) {
    const float* p = sb + ((size_t)g << 3);          // 32B/lane, coalesced
#pragma unroll
    for (int j = 0; j < 8; ++j) { const float v = p[j]; acc[j] += v * v; }
  }
#pragma unroll
  for (int j = 0; j < 8; ++j) red[j * 256 + tid] = acc[j];
  __syncthreads();
  for (int str = 128; str > 0; str >>= 1) {
    if (tid < str) {
#pragma unroll
      for (int j = 0; j < 8; ++j)
        red[j * 256 + tid] += red[j * 256 + tid + str];
    }
    __syncthreads();
  }
  if (tid < 8) out[b * 8 + tid] = red[tid * 256];
}

// ---------------------------------------------------------------------------
extern "C" void kernel_launch(void* const* d_in, const int* in_sizes, int n_in,
                              void* d_out, int out_size, void* d_ws,
                              size_t ws_size, hipStream_t stream) {
  (void)in_sizes; (void)n_in; (void)out_size; (void)ws_size;
  const float* x     = (const float*)d_in[0];  // [8, 2^20] normalized states
  const float* theta = (const float*)d_in[1];  // [13*20]

  // Workspace: two 32 MB ping-pong state buffers + 65 KB of gate matrices.
  float* bufA = (float*)d_ws;
  float* bufB = bufA + ((size_t)BATCH << 20);
  float* AG   = bufB + ((size_t)BATCH << 20);  // 13*5*256 floats

  qae_prep<<<65, 256, 0, stream>>>(theta, AG);

  const size_t lds = TILE_ELEMS * sizeof(float);  // 256 KB dynamic LDS
  for (int t = 0; t <= NLAYERS; ++t) {
    const float* sin_ = (t == 0) ? x : bufB;
    const float* ag   = AG + (size_t)t * 5 * 256;
    // Pass A: (pending CNOT-perm gather for t>0) + groups 1..4
    qae_low<<<BATCH * 16, 256, lds, stream>>>(sin_, bufA, ag, t > 0 ? 1 : 0);
    // Pass B: group 0; the trailing CNOT of layer t (t<12) is deferred to
    // the gather of pass A of layer t+1.
    qae_hi<<<BATCH * 16, 256, lds, stream>>>(bufA, bufB, ag);
  }
  qae_probs<<<BATCH, 256, 0, stream>>>(bufB, (float*)d_out);
}